// HyperConv_77996606095425
// MI455X (gfx1250) — compile-verified
//
#include <hip/hip_runtime.h>

#define D_FEAT 48
#define EPB 64  // edges per block (src/dst/w slices are each exactly 256B)

// ---------------------------------------------------------------------------
// out = feat * scale ; x1 = 0 ; x2 = 0   (float4 vectorized, N*48 % 4 == 0)
// ---------------------------------------------------------------------------
__global__ __launch_bounds__(256) void hc_init(const float4* __restrict__ feat,
                                               float4* __restrict__ out,
                                               float4* __restrict__ x1,
                                               float4* __restrict__ x2,
                                               int n4, float scale) {
    int i = blockIdx.x * blockDim.x + threadIdx.x;
    if (i < n4) {
        float4 f = feat[i];
        out[i] = make_float4(f.x * scale, f.y * scale, f.z * scale, f.w * scale);
        float4 z = make_float4(0.f, 0.f, 0.f, 0.f);
        x1[i] = z;
        x2[i] = z;
    }
}

// ---------------------------------------------------------------------------
// out += x * scale
// ---------------------------------------------------------------------------
__global__ __launch_bounds__(256) void hc_accum(const float4* __restrict__ x,
                                                float4* __restrict__ out,
                                                int n4, float scale) {
    int i = blockIdx.x * blockDim.x + threadIdx.x;
    if (i < n4) {
        float4 v = x[i];
        float4 o = out[i];
        o.x += v.x * scale; o.y += v.y * scale;
        o.z += v.z * scale; o.w += v.w * scale;
        out[i] = o;
    }
}

// ---------------------------------------------------------------------------
// xout[dst[e]] += w[e] * xin[src[e]]   (COO scatter SpMM)
//
// Per block: 64 edges. Waves 0..2 stage src/dst/w (256B each) into LDS via
// the CDNA5 async DMA path (global_load_async_to_lds_b64, ASYNCcnt), then
// 16 lanes x 3 floats cooperate per edge: coalesced 192B gather of the
// source row + 3 global_atomic_add_f32 into the destination row.
// ---------------------------------------------------------------------------
__global__ __launch_bounds__(256) void hc_scatter(const int* __restrict__ src,
                                                  const int* __restrict__ dst,
                                                  const float* __restrict__ w,
                                                  const float* __restrict__ xin,
                                                  float* __restrict__ xout,
                                                  int E) {
    __shared__ int   s_src[EPB];
    __shared__ int   s_dst[EPB];
    __shared__ float s_w[EPB];

    const int tid  = threadIdx.x;
    const int e0   = blockIdx.x * EPB;
    const int nval = min(EPB, E - e0);

    const int wv   = tid >> 5;  // wave id within block
    const int lane = tid & 31;

    if (wv < 3) {
        const void* gbase;
        unsigned    ldsbase;
        if (wv == 0)      { gbase = (const void*)(src + e0); ldsbase = (unsigned)(size_t)(void*)s_src; }
        else if (wv == 1) { gbase = (const void*)(dst + e0); ldsbase = (unsigned)(size_t)(void*)s_dst; }
        else              { gbase = (const void*)(w   + e0); ldsbase = (unsigned)(size_t)(void*)s_w;   }
        if (2 * lane < nval) {
            unsigned voff = lane * 8u;  // one b64 chunk (2 elements) per lane
            asm volatile("global_load_async_to_lds_b64 %0, %1, %2"
                         :
                         : "v"(ldsbase + voff), "v"(voff), "s"(gbase)
                         : "memory");
        }
        asm volatile("s_wait_asynccnt 0x0" ::: "memory");
    }
    __syncthreads();

    const int sub = tid & 15;  // feature sub-group: 3 floats each (16*3 = 48)
    const int eg  = tid >> 4;  // edge slot within a pass: 0..15

#pragma unroll
    for (int p = 0; p < EPB / 16; ++p) {
        int el = p * 16 + eg;
        if (el < nval) {
            int   sN  = s_src[el];
            int   dN  = s_dst[el];
            float wgt = s_w[el];
            const float* xr = xin  + (size_t)sN * D_FEAT + sub * 3;
            float*       xw = xout + (size_t)dN * D_FEAT + sub * 3;
            atomicAdd(&xw[0], wgt * xr[0]);
            atomicAdd(&xw[1], wgt * xr[1]);
            atomicAdd(&xw[2], wgt * xr[2]);
        }
    }
}

// ---------------------------------------------------------------------------
// Inputs: d_in[0] = features [N,48] f32, d_in[1] = edge_weight [E] f32,
//         d_in[2] = edge_index [2,E] int32 (src row then dst row).
// d_out: [N,48] f32.  d_ws: x1 and x2 buffers (2 * N*48*4 = 38.4 MB).
// ---------------------------------------------------------------------------
extern "C" void kernel_launch(void* const* d_in, const int* in_sizes, int n_in,
                              void* d_out, int out_size, void* d_ws, size_t ws_size,
                              hipStream_t stream) {
    const float* feat = (const float*)d_in[0];
    const float* ew   = (const float*)d_in[1];
    const int*   ei   = (const int*)d_in[2];

    const int N = in_sizes[0] / D_FEAT;   // 100000
    const int E = in_sizes[2] / 2;        // 1600000

    float* out = (float*)d_out;
    float* x1  = (float*)d_ws;
    float* x2  = x1 + (size_t)N * D_FEAT;

    const int* src = ei;
    const int* dst = ei + E;

    const float scale = 1.0f / 3.0f;      // 1 / (NUM_LAYERS + 1)
    const int n4      = (N * D_FEAT) / 4;
    const int gElem   = (n4 + 255) / 256;
    const int gEdge   = (E + EPB - 1) / EPB;

    // out = f/3 ; x1 = x2 = 0
    hc_init<<<gElem, 256, 0, stream>>>((const float4*)feat, (float4*)out,
                                       (float4*)x1, (float4*)x2, n4, scale);
    // x1 = A f
    hc_scatter<<<gEdge, 256, 0, stream>>>(src, dst, ew, feat, x1, E);
    // out += x1/3
    hc_accum<<<gElem, 256, 0, stream>>>((const float4*)x1, (float4*)out, n4, scale);
    // x2 = A x1
    hc_scatter<<<gEdge, 256, 0, stream>>>(src, dst, ew, x1, x2, E);
    // out += x2/3
    hc_accum<<<gElem, 256, 0, stream>>>((const float4*)x2, (float4*)out, n4, scale);
}